// MultiHeadAttention_20985210208760
// MI455X (gfx1250) — compile-verified
//
#include <hip/hip_runtime.h>
#include <hip/hip_bf16.h>

// MHA forward for MI455X (gfx1250), fp32 end-to-end:
//   - matrix ops: V_WMMA_F32_16X16X4_F32 (full fp32 precision, matches reference)
//   - tile movement: Tensor Data Mover (tensor_load_to_lds, 6-arg clang-23
//     builtin), double-buffered, tracked with TENSORcnt.  TDM pad_amount
//     reproduces the bank-conflict-avoiding padded LDS strides.
// B=4, S=2048, D=1024, H=16, hd=64.

typedef __attribute__((ext_vector_type(2))) float v2f;
typedef __attribute__((ext_vector_type(8))) float v8f;
typedef __attribute__((ext_vector_type(4))) unsigned int u32x4;
typedef __attribute__((ext_vector_type(8))) int i32x8;
typedef __attribute__((ext_vector_type(4))) int i32x4;

#define B_  4
#define S_  2048
#define D_  1024
#define H_  16
#define HD_ 64

__device__ __forceinline__ v8f wmma_f32(v2f a, v2f b, v8f c) {
  return __builtin_amdgcn_wmma_f32_16x16x4_f32(false, a, false, b, (short)0, c,
                                               false, false);
}

// ---------------------------------------------------------------------------
// TDM: DMA a 2D fp32 tile (tile_d0 elems/row x tile_d1 rows, row stride
// `row_stride` elems) from global to LDS at byte offset lds_off, appending
// pad_dw dwords of LDS padding after every 2^pad_code 8-byte units (== one row
// when the row is 2^pad_code*2 dwords).  Tensor dims == tile dims so no OOB.
// D# layout per CDNA5 ISA.  Whole-wave op: call
// from one wave only; EXEC is ignored by hardware.
// ---------------------------------------------------------------------------
__device__ __forceinline__ void tdm_load_2d(unsigned int lds_off,
                                            const void* gptr,
                                            unsigned int tile_d0,
                                            unsigned int tile_d1,
                                            unsigned long long row_stride,
                                            unsigned int pad_code,
                                            unsigned int pad_dw) {
  const unsigned long long ga = (unsigned long long)(uintptr_t)gptr;
  u32x4 g0;
  g0[0] = 1u;                                        // count=1 (valid), user mode
  g0[1] = lds_off;                                   // lds_addr [63:32]
  g0[2] = (unsigned int)ga;                          // global_addr [95:64]
  g0[3] = (unsigned int)((ga >> 32) & 0x1FFFFFFu)    // global_addr [120:96]
          | (2u << 30);                              // type=2 ("image")
  i32x8 g1;
  g1[0] = (int)((2u << 16)                           // data_size = 4B
                | (1u << 20)                         // pad_enable
                | (pad_code << 22)                   // pad_interval
                | ((pad_dw - 1u) << 25));            // pad_amount (0 => 1 dw)
  g1[1] = (int)((tile_d0 & 0xFFFFu) << 16);          // tensor_dim0[15:0]
  g1[2] = (int)((tile_d0 >> 16) | ((tile_d1 & 0xFFFFu) << 16));  // d0 hi | d1 lo
  g1[3] = (int)((tile_d1 >> 16) | ((tile_d0 & 0xFFFFu) << 16));  // d1 hi | tile_dim0
  g1[4] = (int)(tile_d1 & 0xFFFFu);                  // tile_dim1 | tile_dim2=0
  g1[5] = (int)(unsigned int)(row_stride);           // tensor_dim0_stride[31:0]
  g1[6] = (int)(unsigned int)((row_stride >> 32) & 0xFFFFu);  // stride[47:32]
  g1[7] = 0;                                         // tensor_dim1_stride = 0 (2D)
  const i32x4 z4 = {0, 0, 0, 0};                     // groups 2/3 unused (2D)
  const i32x8 z8 = {0, 0, 0, 0, 0, 0, 0, 0};         // trailing group (clang-23 form)
  __builtin_amdgcn_tensor_load_to_lds(g0, g1, z4, z4, z8, 0);
}

__device__ __forceinline__ unsigned int lds_off(const void* p) {
  // generic shared pointer = SHARED_BASE aperture | 32-bit LDS offset (ISA §10.2)
  return (unsigned int)(uintptr_t)p;
}

// ---------------------------------------------------------------------------
// Kernel 1/5: C[M,N] = X[M,K] @ W[K,N] + bias[N]
// PERMUTE=1 -> store into [B,H,S,hd]; PERMUTE=0 -> [M,N].
// 128 threads = 4 waves; 64x64 C tile; K staged 32 wide, TDM double-buffered.
// ---------------------------------------------------------------------------
template <int PERMUTE>
__global__ __launch_bounds__(128) void mha_gemm_proj(
    const float* __restrict__ X, const float* __restrict__ W,
    const float* __restrict__ bias, float* __restrict__ Out) {
  __shared__ float lA[2][64][36];  // 64 rows x 32 k, +4 dw pad/row (TDM-written)
  __shared__ float lB[2][32][68];  // 32 k x 64 n, +4 dw pad/row

  const int tid = threadIdx.x;
  const int wave = tid >> 5;
  const int lane = tid & 31;
  const int row16 = lane & 15;   // N index (C/D layout)
  const int khalf = lane >> 4;   // K-block select (A/B) / M+8 (C/D)
  const int n0 = blockIdx.x * 64;
  const int m0 = blockIdx.y * 64;

  if (wave == 0) {  // prologue: stage 0 into buffer 0
    tdm_load_2d(lds_off(&lA[0][0][0]), X + (size_t)m0 * D_, 32, 64, D_, 4, 4);
    tdm_load_2d(lds_off(&lB[0][0][0]), W + n0, 64, 32, D_, 5, 4);
  }

  v8f acc[4] = {};
  int buf = 0;
  for (int kt = 0; kt < D_; kt += 32) {
    if (wave == 0) {
      if (kt + 32 < D_) {  // issue next stage, then wait for current (<=2 left)
        tdm_load_2d(lds_off(&lA[buf ^ 1][0][0]),
                    X + (size_t)m0 * D_ + (kt + 32), 32, 64, D_, 4, 4);
        tdm_load_2d(lds_off(&lB[buf ^ 1][0][0]),
                    W + (size_t)(kt + 32) * D_ + n0, 64, 32, D_, 5, 4);
        __builtin_amdgcn_s_wait_tensorcnt((short)2);
      } else {
        __builtin_amdgcn_s_wait_tensorcnt((short)0);
      }
    }
    __syncthreads();

#pragma unroll
    for (int kk = 0; kk < 32; kk += 4) {
      v2f a;  // A 16x4 f32: lanes 0-15 -> K0/K1, lanes 16-31 -> K2/K3
      a.x = lA[buf][wave * 16 + row16][kk + 2 * khalf + 0];
      a.y = lA[buf][wave * 16 + row16][kk + 2 * khalf + 1];
#pragma unroll
      for (int t = 0; t < 4; ++t) {
        v2f b;
        b.x = lB[buf][kk + 2 * khalf + 0][t * 16 + row16];
        b.y = lB[buf][kk + 2 * khalf + 1][t * 16 + row16];
        acc[t] = wmma_f32(a, b, acc[t]);
      }
    }
    __syncthreads();  // all reads of buf done before it is refilled next iter
    buf ^= 1;
  }

#pragma unroll
  for (int t = 0; t < 4; ++t) {
    const int n = n0 + t * 16 + row16;
    const float bv = bias[n];
#pragma unroll
    for (int r = 0; r < 8; ++r) {
      const int m = m0 + wave * 16 + r + 8 * khalf;
      const float val = acc[t][r] + bv;
      if (PERMUTE) {  // -> [B,H,S,hd]
        const int bb = m >> 11, ss = m & (S_ - 1);
        const int h = n >> 6, d = n & (HD_ - 1);
        Out[(((size_t)bb * H_ + h) * S_ + ss) * HD_ + d] = val;
      } else {
        Out[(size_t)m * D_ + n] = val;
      }
    }
  }
}

// ---------------------------------------------------------------------------
// Kernel 2/5: scores[bh,i,j] = (Q[bh,i,:] . K[bh,j,:]) / sqrt(hd)
// hd=64 => single TDM stage; block = 64(i) x 64(j), 4 waves.
// ---------------------------------------------------------------------------
__global__ __launch_bounds__(128) void mha_scores(
    const float* __restrict__ Q, const float* __restrict__ K,
    float* __restrict__ scores) {
  __shared__ float lQ[64][68];
  __shared__ float lK[64][68];

  const int tid = threadIdx.x;
  const int wave = tid >> 5;
  const int lane = tid & 31;
  const int row16 = lane & 15;
  const int khalf = lane >> 4;
  const int j0 = blockIdx.x * 64;
  const int i0 = blockIdx.y * 64;
  const int bh = blockIdx.z;
  const float* Qh = Q + (size_t)bh * S_ * HD_;
  const float* Kh = K + (size_t)bh * S_ * HD_;

  if (wave == 0) {
    tdm_load_2d(lds_off(&lQ[0][0]), Qh + (size_t)i0 * HD_, 64, 64, HD_, 5, 4);
    tdm_load_2d(lds_off(&lK[0][0]), Kh + (size_t)j0 * HD_, 64, 64, HD_, 5, 4);
    __builtin_amdgcn_s_wait_tensorcnt((short)0);
  }
  __syncthreads();

  v8f acc[4] = {};
#pragma unroll
  for (int kk = 0; kk < HD_; kk += 4) {
    v2f a;
    a.x = lQ[wave * 16 + row16][kk + 2 * khalf + 0];
    a.y = lQ[wave * 16 + row16][kk + 2 * khalf + 1];
#pragma unroll
    for (int t = 0; t < 4; ++t) {
      v2f b;  // B[d][j] = K[j][d] via LDS transpose read
      b.x = lK[t * 16 + row16][kk + 2 * khalf + 0];
      b.y = lK[t * 16 + row16][kk + 2 * khalf + 1];
      acc[t] = wmma_f32(a, b, acc[t]);
    }
  }

  const float scale = 0.125f;  // 1/sqrt(64)
#pragma unroll
  for (int t = 0; t < 4; ++t) {
    const int j = j0 + t * 16 + row16;
#pragma unroll
    for (int r = 0; r < 8; ++r) {
      const int i = i0 + wave * 16 + r + 8 * khalf;
      scores[((size_t)bh * S_ + i) * S_ + j] = acc[t][r] * scale;
    }
  }
}

// ---------------------------------------------------------------------------
// Kernel 3/5: masked softmax over each row of [B,H,S, S_]; in place.
// Fully-masked row -> zeros (matches reference NaN cleanup).
// ---------------------------------------------------------------------------
__global__ __launch_bounds__(256) void mha_softmax(
    float* __restrict__ attn, const unsigned char* __restrict__ mask) {
  __shared__ float sdata[256];
  const int row = blockIdx.x;  // = bh*S + i
  const int b = row >> 15;     // / (H_*S_)
  float* p = attn + (size_t)row * S_;
  const unsigned char* mrow = mask + (size_t)b * S_;
  const int tid = threadIdx.x;

  float v[8];
  float mx = -INFINITY;
#pragma unroll
  for (int i = 0; i < 8; ++i) {
    const int j = tid + (i << 8);
    float x = p[j];
    x = mrow[j] ? x : -INFINITY;
    v[i] = x;
    mx = fmaxf(mx, x);
  }
  sdata[tid] = mx;
  __syncthreads();
  for (int s = 128; s > 0; s >>= 1) {
    if (tid < s) sdata[tid] = fmaxf(sdata[tid], sdata[tid + s]);
    __syncthreads();
  }
  mx = sdata[0];
  __syncthreads();

  float sum = 0.f;
#pragma unroll
  for (int i = 0; i < 8; ++i) {
    const float e = (v[i] == -INFINITY) ? 0.f : __expf(v[i] - mx);
    v[i] = e;
    sum += e;
  }
  sdata[tid] = sum;
  __syncthreads();
  for (int s = 128; s > 0; s >>= 1) {
    if (tid < s) sdata[tid] += sdata[tid + s];
    __syncthreads();
  }
  const float inv = sdata[0] > 0.f ? 1.0f / sdata[0] : 0.f;
#pragma unroll
  for (int i = 0; i < 8; ++i) p[tid + (i << 8)] = v[i] * inv;
}

// ---------------------------------------------------------------------------
// Kernel 4/5: Oh[b,s, h*64+d] = sum_j attn[bh,i,j] * V[bh,j,d]
// Block: 64(i) x 64(d = full head); k over S in 32-wide TDM double-buffered
// stages.  This kernel streams the 1 GB attn tensor -> keep VMEM saturated.
// ---------------------------------------------------------------------------
__global__ __launch_bounds__(128) void mha_attnv(
    const float* __restrict__ attn, const float* __restrict__ V,
    float* __restrict__ Oh) {
  __shared__ float lA[2][64][36];
  __shared__ float lV[2][32][68];

  const int tid = threadIdx.x;
  const int wave = tid >> 5;
  const int lane = tid & 31;
  const int row16 = lane & 15;
  const int khalf = lane >> 4;
  const int i0 = blockIdx.y * 64;
  const int bh = blockIdx.z;
  const int b = bh >> 4, h = bh & 15;
  const float* Ah = attn + (size_t)bh * S_ * S_;
  const float* Vh = V + (size_t)bh * S_ * HD_;

  if (wave == 0) {
    tdm_load_2d(lds_off(&lA[0][0][0]), Ah + (size_t)i0 * S_, 32, 64, S_, 4, 4);
    tdm_load_2d(lds_off(&lV[0][0][0]), Vh, 64, 32, HD_, 5, 4);
  }

  v8f acc[4] = {};
  int buf = 0;
  for (int jt = 0; jt < S_; jt += 32) {
    if (wave == 0) {
      if (jt + 32 < S_) {
        tdm_load_2d(lds_off(&lA[buf ^ 1][0][0]),
                    Ah + (size_t)i0 * S_ + (jt + 32), 32, 64, S_, 4, 4);
        tdm_load_2d(lds_off(&lV[buf ^ 1][0][0]),
                    Vh + (size_t)(jt + 32) * HD_, 64, 32, HD_, 5, 4);
        __builtin_amdgcn_s_wait_tensorcnt((short)2);
      } else {
        __builtin_amdgcn_s_wait_tensorcnt((short)0);
      }
    }
    __syncthreads();

#pragma unroll
    for (int kk = 0; kk < 32; kk += 4) {
      v2f a;
      a.x = lA[buf][wave * 16 + row16][kk + 2 * khalf + 0];
      a.y = lA[buf][wave * 16 + row16][kk + 2 * khalf + 1];
#pragma unroll
      for (int t = 0; t < 4; ++t) {
        v2f bb;
        bb.x = lV[buf][kk + 2 * khalf + 0][t * 16 + row16];
        bb.y = lV[buf][kk + 2 * khalf + 1][t * 16 + row16];
        acc[t] = wmma_f32(a, bb, acc[t]);
      }
    }
    __syncthreads();
    buf ^= 1;
  }

#pragma unroll
  for (int t = 0; t < 4; ++t) {
    const int d = t * 16 + row16;
#pragma unroll
    for (int r = 0; r < 8; ++r) {
      const int i = i0 + wave * 16 + r + 8 * khalf;
      Oh[((size_t)b * S_ + i) * D_ + h * HD_ + d] = acc[t][r];
    }
  }
}

// ---------------------------------------------------------------------------
extern "C" void kernel_launch(void* const* d_in, const int* in_sizes, int n_in,
                              void* d_out, int out_size, void* d_ws,
                              size_t ws_size, hipStream_t stream) {
  const float* query = (const float*)d_in[0];
  const float* key_  = (const float*)d_in[1];
  const float* value = (const float*)d_in[2];
  const unsigned char* mask = (const unsigned char*)d_in[3];  // jax bool = 1B
  const float* w_q = (const float*)d_in[4];
  const float* b_q = (const float*)d_in[5];
  const float* w_k = (const float*)d_in[6];
  const float* b_k = (const float*)d_in[7];
  const float* w_v = (const float*)d_in[8];
  const float* b_v = (const float*)d_in[9];
  const float* w_o = (const float*)d_in[10];
  const float* b_o = (const float*)d_in[11];

  const size_t act = (size_t)B_ * S_ * D_;  // 8,388,608 floats
  float* out  = (float*)d_out;              // [B,S,D]
  float* attn = out + act;                  // [B,H,S,S]

  float* Q  = (float*)d_ws;                 // [B,H,S,hd]
  float* K  = Q + act;
  float* V  = K + act;
  float* Oh = Q;                            // reuse Q region (dead after scores)

  const dim3 gp(D_ / 64, (B_ * S_) / 64);   // (16,128)
  mha_gemm_proj<1><<<gp, 128, 0, stream>>>(query, w_q, b_q, Q);
  mha_gemm_proj<1><<<gp, 128, 0, stream>>>(key_,  w_k, b_k, K);
  mha_gemm_proj<1><<<gp, 128, 0, stream>>>(value, w_v, b_v, V);

  const dim3 gs(S_ / 64, S_ / 64, B_ * H_); // (32,32,64)
  mha_scores<<<gs, 128, 0, stream>>>(Q, K, attn);

  mha_softmax<<<dim3(B_ * H_ * S_), 256, 0, stream>>>(attn, mask);

  const dim3 gv(1, S_ / 64, B_ * H_);
  mha_attnv<<<gv, 128, 0, stream>>>(attn, V, Oh);

  mha_gemm_proj<0><<<gp, 128, 0, stream>>>(Oh, w_o, b_o, out);
}